// GNN_33062658245508
// MI455X (gfx1250) — compile-verified
//
#include <hip/hip_runtime.h>

#define N_NODES 100000
#define N_EDGES 1600000
#define IN_CH   128
#define HID_CH  256
#define OUT_CH  128

typedef __attribute__((ext_vector_type(2))) float v2f;
typedef __attribute__((ext_vector_type(8))) float v8f;

// ---------------------------------------------------------------- zero
__global__ void GNN_zero_f32(float* __restrict__ p, long n4) {
    long i = (long)blockIdx.x * blockDim.x + threadIdx.x;
    long stride = (long)gridDim.x * blockDim.x;
    float4* p4 = reinterpret_cast<float4*>(p);
    for (; i < n4; i += stride) p4[i] = make_float4(0.f, 0.f, 0.f, 0.f);
}

// ---------------------------------------------------------------- degree
__global__ void GNN_degree(const long long* __restrict__ ei, float* __restrict__ deg) {
    int e = blockIdx.x * blockDim.x + threadIdx.x;   // exactly N_EDGES threads
    int d = (int)ei[N_EDGES + e];
    unsafeAtomicAdd(&deg[d], 1.0f);
}

__global__ void GNN_invdeg(float* __restrict__ deg) {
    int i = blockIdx.x * blockDim.x + threadIdx.x;
    if (i < N_NODES) deg[i] = 1.0f / fmaxf(deg[i], 1.0f);
}

// ---------------------------------------------------------------- edge scatter
// blockDim.x = C/4 channel chunks (coalesced row gather), blockDim.y edges/block
template <int C>
__global__ void GNN_scatter(const float* __restrict__ x,
                            const long long* __restrict__ ei,
                            float* __restrict__ agg) {
    int e = blockIdx.x * blockDim.y + threadIdx.y;   // grids sized to divide exactly
    int s = (int)ei[e];
    int d = (int)ei[N_EDGES + e];
    int c = threadIdx.x * 4;
    const float4 v = *reinterpret_cast<const float4*>(&x[(long)s * C + c]);
    float* dst = &agg[(long)d * C + c];
    unsafeAtomicAdd(dst + 0, v.x);
    unsafeAtomicAdd(dst + 1, v.y);
    unsafeAtomicAdd(dst + 2, v.z);
    unsafeAtomicAdd(dst + 3, v.w);
}

// ---------------------------------------------------------------- weight repack
// Repack row-major W[K][N] into WMMA-native B pairs so a wave's B fetch is one
// coalesced b64 per lane:
//   Wp[((g*(N/16) + t)*2 + h)*16 + nlo] = { W[4g+2h][16t+nlo], W[4g+2h+1][16t+nlo] }
template <int K, int N>
__global__ void GNN_pack_w(const float* __restrict__ W, v2f* __restrict__ Wp) {
    int i = blockIdx.x * blockDim.x + threadIdx.x;   // over K*N/2 pairs (exact grid)
    int nlo = i & 15;
    int h   = (i >> 4) & 1;
    int t   = (i >> 5) % (N / 16);
    int g   = i / ((N / 16) * 32);
    int k   = 4 * g + 2 * h;
    int n   = 16 * t + nlo;
    v2f val;
    val.x = W[(long)k * N + n];
    val.y = W[(long)(k + 1) * N + n];
    Wp[i] = val;
}

// ---------------------------------------------------------------- fused SAGE linear
// out = relu?( (agg*invdeg) @ Wl + bias + x @ Wr ), f32 end-to-end.
// One wave -> 32x32 output (2M x 2N tiles, 4 accumulators); per k-step:
// 2 A b64 loads + 2 packed-B b64 loads feed 4 x V_WMMA_F32_16X16X4_F32.
template <int K, int N, bool SCALE>
__device__ __forceinline__ void gemm_phase(const float* __restrict__ ArowA,
                                           const float* __restrict__ ArowB,
                                           const v2f* __restrict__ Wp,
                                           float sA, float sB,
                                           int t0, int hi2, int nlo,
                                           v8f& acc00, v8f& acc01,
                                           v8f& acc10, v8f& acc11) {
    for (int k = 0; k < K; k += 4) {
        const int g  = k >> 2;
        const int ka = k + 2 * hi2;
        v2f a0 = *(const v2f*)(ArowA + ka);
        v2f a1 = *(const v2f*)(ArowB + ka);
        if (SCALE) { a0 *= sA; a1 *= sB; }
        const long base = ((long)g * (N / 16) + t0) * 32 + hi2 * 16 + nlo;
        const v2f b0 = Wp[base];
        const v2f b1 = Wp[base + 32];
        acc00 = __builtin_amdgcn_wmma_f32_16x16x4_f32(false, a0, false, b0, (short)0, acc00, false, false);
        acc01 = __builtin_amdgcn_wmma_f32_16x16x4_f32(false, a0, false, b1, (short)0, acc01, false, false);
        acc10 = __builtin_amdgcn_wmma_f32_16x16x4_f32(false, a1, false, b0, (short)0, acc10, false, false);
        acc11 = __builtin_amdgcn_wmma_f32_16x16x4_f32(false, a1, false, b1, (short)0, acc11, false, false);
    }
}

template <int K, int N, bool RELU>
__global__ void GNN_sage_gemm(const float* __restrict__ agg,
                              const float* __restrict__ invdeg,
                              const float* __restrict__ x,
                              const v2f* __restrict__ Wlp,
                              const v2f* __restrict__ Wrp,
                              const float* __restrict__ bias,
                              float* __restrict__ out) {
    const int lane = threadIdx.x;                     // 0..31
    const int nlo  = lane & 15;
    const int hi2  = lane >> 4;                       // 0 or 1
    const int np   = blockIdx.x * blockDim.y + threadIdx.y;   // N-pair index
    const int n0   = np * 32;
    const int t0   = n0 / 16;
    const int m0   = blockIdx.y * 32;
    const int mA   = m0 + nlo;
    const int mB   = mA + 16;

    const float bv0 = bias[n0 + nlo];
    const float bv1 = bias[n0 + 16 + nlo];
    v8f acc00, acc01, acc10, acc11;
#pragma unroll
    for (int i = 0; i < 8; ++i) {
        acc00[i] = bv0; acc01[i] = bv1; acc10[i] = bv0; acc11[i] = bv1;
    }

    const float sA = invdeg[mA];
    const float sB = invdeg[mB];

    // phase 0: mean = agg * invdeg   (scaled)
    gemm_phase<K, N, true >(agg + (long)mA * K, agg + (long)mB * K, Wlp,
                            sA, sB, t0, hi2, nlo, acc00, acc01, acc10, acc11);
    // phase 1: + x @ Wr              (unscaled)
    gemm_phase<K, N, false>(x + (long)mA * K, x + (long)mB * K, Wrp,
                            0.f, 0.f, t0, hi2, nlo, acc00, acc01, acc10, acc11);

#pragma unroll
    for (int i = 0; i < 8; ++i) {
        float v00 = acc00[i], v01 = acc01[i], v10 = acc10[i], v11 = acc11[i];
        if (RELU) {
            v00 = fmaxf(v00, 0.f); v01 = fmaxf(v01, 0.f);
            v10 = fmaxf(v10, 0.f); v11 = fmaxf(v11, 0.f);
        }
        const int r0 = m0 + i + 8 * hi2;
        const int r1 = r0 + 16;
        out[(long)r0 * N + n0 + nlo]      = v00;
        out[(long)r0 * N + n0 + 16 + nlo] = v01;
        out[(long)r1 * N + n0 + nlo]      = v10;
        out[(long)r1 * N + n0 + 16 + nlo] = v11;
    }
}

// ---------------------------------------------------------------- launcher
extern "C" void kernel_launch(void* const* d_in, const int* in_sizes, int n_in,
                              void* d_out, int out_size, void* d_ws, size_t ws_size,
                              hipStream_t stream) {
    const float*     x   = (const float*)d_in[0];
    const long long* ei  = (const long long*)d_in[1];
    const float*     W1l = (const float*)d_in[2];
    const float*     b1  = (const float*)d_in[3];
    const float*     W1r = (const float*)d_in[4];
    const float*     W2l = (const float*)d_in[5];
    const float*     b2  = (const float*)d_in[6];
    const float*     W2r = (const float*)d_in[7];
    float*           out = (float*)d_out;

    // workspace layout (floats): deg | agg1 | agg2 | h | packed weights
    float* ws    = (float*)d_ws;
    float* deg   = ws;                                       // 100,000
    float* agg1  = ws + 100000;                              // N*128
    float* agg2  = agg1 + (long)N_NODES * IN_CH;             // N*256
    float* h     = agg2 + (long)N_NODES * HID_CH;            // N*256
    v2f*   W1lp  = (v2f*)(h + (long)N_NODES * HID_CH);       // 128*256 floats
    v2f*   W1rp  = W1lp + (IN_CH  * HID_CH / 2);
    v2f*   W2lp  = W1rp + (IN_CH  * HID_CH / 2);
    v2f*   W2rp  = W2lp + (HID_CH * OUT_CH / 2);

    // 1) zero deg+agg1+agg2 (contiguous): 38,500,000 floats = 9,625,000 float4
    {
        long n4 = (100000L + (long)N_NODES * IN_CH + (long)N_NODES * HID_CH) / 4;
        GNN_zero_f32<<<4096, 256, 0, stream>>>(ws, n4);
    }
    // 2) repack weights into WMMA B layout (tiny; overlaps nothing it needs)
    GNN_pack_w<IN_CH,  HID_CH><<<IN_CH  * HID_CH / 2 / 256, 256, 0, stream>>>(W1l, W1lp);
    GNN_pack_w<IN_CH,  HID_CH><<<IN_CH  * HID_CH / 2 / 256, 256, 0, stream>>>(W1r, W1rp);
    GNN_pack_w<HID_CH, OUT_CH><<<HID_CH * OUT_CH / 2 / 256, 256, 0, stream>>>(W2l, W2lp);
    GNN_pack_w<HID_CH, OUT_CH><<<HID_CH * OUT_CH / 2 / 256, 256, 0, stream>>>(W2r, W2rp);
    // 3) degree + scatter layer 1
    GNN_degree<<<N_EDGES / 256, 256, 0, stream>>>(ei, deg);
    {
        dim3 blk(IN_CH / 4, 8);                              // 32 x 8 = 256
        GNN_scatter<IN_CH><<<N_EDGES / 8, blk, 0, stream>>>(x, ei, agg1);
    }
    // 4) invert degrees (shared by both layers)
    GNN_invdeg<<<(N_NODES + 255) / 256, 256, 0, stream>>>(deg);
    // 5) layer-1 fused GEMM + bias + ReLU -> h  [M=100000, K=128, N=256]
    {
        dim3 blk(32, 4);                                     // 4 waves / block
        dim3 grd((HID_CH / 32) / 4, N_NODES / 32);           // (2, 3125)
        GNN_sage_gemm<IN_CH, HID_CH, true>
            <<<grd, blk, 0, stream>>>(agg1, deg, x, W1lp, W1rp, b1, h);
    }
    // 6) scatter layer 2 on h
    {
        dim3 blk(HID_CH / 4, 4);                             // 64 x 4 = 256
        GNN_scatter<HID_CH><<<N_EDGES / 4, blk, 0, stream>>>(h, ei, agg2);
    }
    // 7) layer-2 fused GEMM + bias -> out  [M=100000, K=256, N=128]
    {
        dim3 blk(32, 4);
        dim3 grd((OUT_CH / 32) / 4, N_NODES / 32);           // (1, 3125)
        GNN_sage_gemm<HID_CH, OUT_CH, false>
            <<<grd, blk, 0, stream>>>(agg2, deg, h, W2lp, W2rp, b2, out);
    }
}